// QuestionAnswerModel_28140625723971
// MI455X (gfx1250) — compile-verified
//
#include <hip/hip_runtime.h>
#include <hip/hip_bf16.h>

typedef __attribute__((ext_vector_type(2))) float v2f;
typedef __attribute__((ext_vector_type(4))) float v4f;
typedef __attribute__((ext_vector_type(8))) float v8f;

#define PI_F 3.1415926535f
#define Bn   256
#define Dn   256
#define Hn   768
#define ENTn 43234

// ---------------------------------------------------------------------------
// Kernel 1: rel = relu(q) @ W^T + b via V_WMMA_F32_16X16X4_F32, fused RotatE
// epilogue producing head_sum[b] = sum_d(|re|+|im|).
//
// Grid: 16 blocks (one per 16-row stripe of B=256), 512 threads = 16 waves.
// Wave w computes the 16x16 tile at columns [16w, 16w+16).
//
// A-frag (16x4 f32, 2 VGPRs): lane L -> row m = L&15, k = kbase + 2*(L>>4)+j
// B-frag (4x16 f32, 2 VGPRs): lane L -> col n = L&15, same k mapping
// C/D   (16x16 f32, 8 VGPRs): lane L -> col n = L&15, row m = v + 8*(L>>4)
// ---------------------------------------------------------------------------
__global__ __launch_bounds__(512)
void qa_gemm_head_kernel(const float* __restrict__ q,
                         const int*   __restrict__ head_id,
                         const float* __restrict__ W,
                         const float* __restrict__ bias,
                         const float* __restrict__ ent,
                         float*       __restrict__ head_sum) {
    __shared__ float lds_hs[16];

    const int lane    = threadIdx.x & 31;
    const int wave    = threadIdx.x >> 5;       // 0..15 -> column tile
    const int half    = lane >> 4;              // 0 or 1
    const int l15     = lane & 15;
    const int rowBase = blockIdx.x * 16;
    const int colBase = wave * 16;

    if (threadIdx.x < 16) lds_hs[threadIdx.x] = 0.0f;
    __syncthreads();

    // Per-lane base pointers (8-byte aligned: row*768 and 2*half are even).
    const float* arow = q + (size_t)(rowBase + l15) * Hn + 2 * half;
    const float* brow = W + (size_t)(colBase + l15) * Hn + 2 * half;

    v8f acc = {};
    for (int k = 0; k < Hn; k += 4) {
        v2f a = *(const v2f*)(arow + k);
        v2f bf = *(const v2f*)(brow + k);
        // ReLU on the A operand
        a.x = a.x > 0.0f ? a.x : 0.0f;
        a.y = a.y > 0.0f ? a.y : 0.0f;
        acc = __builtin_amdgcn_wmma_f32_16x16x4_f32(
                  /*neg_a=*/false, a, /*neg_b=*/false, bf,
                  /*c_mod=*/(short)0, acc, /*reuse_a=*/false, /*reuse_b=*/false);
    }

    // Fused RotatE epilogue: per element (row m, col d), gather head entity,
    // rotate, abs, then reduce over d (lanes within each 16-lane half).
    const int   col  = colBase + l15;
    const float bcol = bias[col];
#pragma unroll
    for (int v = 0; v < 8; ++v) {
        const int m    = v + 8 * half;          // row within stripe
        const int brow_g = rowBase + m;         // global batch row
        float rel = acc[v] + bcol;
        float s, c;
        __sincosf(rel * PI_F, &s, &c);
        const int    hid = head_id[brow_g];
        const float* he  = ent + (size_t)hid * (2 * Dn);
        const float  reh = he[col];
        const float  imh = he[Dn + col];
        float contrib = fabsf(reh * c - imh * s) + fabsf(reh * s + imh * c);
        // Sum over the 16 columns of this tile (stays within the lane half).
        contrib += __shfl_xor(contrib, 1, 32);
        contrib += __shfl_xor(contrib, 2, 32);
        contrib += __shfl_xor(contrib, 4, 32);
        contrib += __shfl_xor(contrib, 8, 32);
        if (l15 == 0) atomicAdd(&lds_hs[m], contrib);  // ds_add_f32
    }
    __syncthreads();
    if (threadIdx.x < 16) head_sum[rowBase + threadIdx.x] = lds_hs[threadIdx.x];
}

// ---------------------------------------------------------------------------
// Kernel 2: tail_sum[e] = sum of all 512 floats of ent_emb row e.
// One wave per row; lane reads 4x float4 (coalesced b128 loads) -> 88.5 MB
// streaming read, the dominant read traffic. 8 waves / block.
// ---------------------------------------------------------------------------
__global__ __launch_bounds__(256)
void qa_tail_sum_kernel(const float* __restrict__ ent,
                        float*       __restrict__ tail_sum) {
    const int wave = threadIdx.x >> 5;
    const int lane = threadIdx.x & 31;
    const int e    = blockIdx.x * 8 + wave;
    if (e >= ENTn) return;

    const float* row = ent + (size_t)e * (2 * Dn);
    float s = 0.0f;
#pragma unroll
    for (int chunk = 0; chunk < 4; ++chunk) {
        v4f d = *(const v4f*)(row + chunk * 128 + lane * 4);
        s += (d.x + d.y) + (d.z + d.w);
    }
    s += __shfl_xor(s, 1, 32);
    s += __shfl_xor(s, 2, 32);
    s += __shfl_xor(s, 4, 32);
    s += __shfl_xor(s, 8, 32);
    s += __shfl_xor(s, 16, 32);
    if (lane == 0) tail_sum[e] = s;
}

// ---------------------------------------------------------------------------
// Kernel 3: score[b,e] = 1 / (head_sum[b] - tail_sum[e]).
// 44.3 MB of coalesced stores; head_sum[b] is block-uniform (scalar load),
// tail_sum (169 KB) stays resident in L2 across the 256 row-passes.
// ---------------------------------------------------------------------------
__global__ __launch_bounds__(256)
void qa_score_kernel(const float* __restrict__ head_sum,
                     const float* __restrict__ tail_sum,
                     float*       __restrict__ out) {
    const int e    = blockIdx.x * 256 + threadIdx.x;
    const int brow = blockIdx.y;
    if (e < ENTn) {
        const float hs = head_sum[brow];
        out[(size_t)brow * ENTn + e] = 1.0f / (hs - tail_sum[e]);
    }
}

extern "C" void kernel_launch(void* const* d_in, const int* in_sizes, int n_in,
                              void* d_out, int out_size, void* d_ws, size_t ws_size,
                              hipStream_t stream) {
    const float* q       = (const float*)d_in[0];  // (256, 768)
    const int*   head_id = (const int*)  d_in[1];  // (256,)
    const float* W       = (const float*)d_in[2];  // (256, 768)
    const float* bias    = (const float*)d_in[3];  // (256,)
    const float* ent     = (const float*)d_in[4];  // (43234, 512)
    float*       out     = (float*)d_out;          // (256, 43234)

    float* head_sum = (float*)d_ws;                // 256 floats
    float* tail_sum = head_sum + Bn;               // 43234 floats

    qa_gemm_head_kernel<<<Bn / 16, 512, 0, stream>>>(q, head_id, W, bias, ent,
                                                     head_sum);
    qa_tail_sum_kernel<<<(ENTn + 7) / 8, 256, 0, stream>>>(ent, tail_sum);

    dim3 sgrid((ENTn + 255) / 256, Bn);
    qa_score_kernel<<<sgrid, 256, 0, stream>>>(head_sum, tail_sum, out);
}